// PAFALoss_37168646980341
// MI455X (gfx1250) — compile-verified
//
#include <hip/hip_runtime.h>
#include <hip/hip_bf16.h>

typedef float v2f __attribute__((ext_vector_type(2)));
typedef float v8f __attribute__((ext_vector_type(8)));

#define NROWS 262144
#define DDIM  256
#define PNUM  512
#define CT    8                 // column tiles
#define COLS  32                // columns per tile (CT*COLS == DDIM)
#define RB    32                // row blocks
#define ROWS_PER_RB (NROWS / RB)        // 8192
#define TPB_B 1024
#define TPR   (COLS / 4)                // 8 threads per row (float4 each)
#define ROWS_PER_SLAB (TPB_B / TPR)     // 128
#define SLABS 4
#define ROWS_PER_ITER (ROWS_PER_SLAB * SLABS)   // 512
#define ITERS (ROWS_PER_RB / ROWS_PER_ITER)     // 16

// ---------------- workspace layout (bytes) ----------------
#define WS_SUMSQ     0          // double
#define WS_S1        8          // float
#define WS_S2        12         // float
#define WS_COUNTS    64         // float[512]
#define WS_CENTROIDS 4096       // float[512][256]
#define WS_CPART     (WS_CENTROIDS + PNUM*DDIM*4)        // float[RB][512]
#define WS_SPART     (WS_CPART + RB*PNUM*4)              // float[CT][RB][512][COLS]

// ---------------- kernel A: zero the scalar accumulators ----------------
__global__ void pafa_init(double* sumsq, float* S1, float* S2) {
    if (threadIdx.x == 0) { *sumsq = 0.0; *S1 = 0.0f; *S2 = 0.0f; }
}

// ---------------- kernel B: streaming segment-sum pass ----------------
__global__ __launch_bounds__(TPB_B) void pafa_partial(
    const float* __restrict__ feat, const int* __restrict__ pid,
    float* __restrict__ spart, float* __restrict__ cpart, double* __restrict__ sumsq)
{
    __shared__ float part[PNUM * COLS];     // 64 KB segment-sum tile
    __shared__ float cnt[PNUM];
    __shared__ int   spid[ROWS_PER_ITER];
    __shared__ float red[TPB_B / 32];

    const int t  = threadIdx.x;
    const int rb = blockIdx.x;    // 0..RB-1   (contiguous 8192-row slice)
    const int ct = blockIdx.y;    // 0..CT-1   (32-column slice)

    for (int i = t; i < PNUM * COLS; i += TPB_B) part[i] = 0.0f;
    if (t < PNUM) cnt[t] = 0.0f;

    const int colBase  = ct * COLS;
    const int rowBase0 = rb * ROWS_PER_RB;
    const int rowInSlab = t >> 3;   // t / TPR
    const int colIdx    = t & 7;    // t % TPR
    float ssq = 0.0f;

    for (int it = 0; it < ITERS; ++it) {
        const int rowBase = rowBase0 + it * ROWS_PER_ITER;
        __syncthreads();                       // protect spid reuse + first-iter LDS zero
        if (t < ROWS_PER_ITER) spid[t] = pid[rowBase + t];
        __syncthreads();
        #pragma unroll
        for (int s = 0; s < SLABS; ++s) {
            const int rloc = s * ROWS_PER_SLAB + rowInSlab;
            const int row  = rowBase + rloc;
            const float4 v = *(const float4*)(feat + (size_t)row * DDIM + colBase + colIdx * 4);
            const int p = spid[rloc];
            float* dst = part + p * COLS + colIdx * 4;
            atomicAdd(dst + 0, v.x);           // ds_add_f32
            atomicAdd(dst + 1, v.y);
            atomicAdd(dst + 2, v.z);
            atomicAdd(dst + 3, v.w);
            ssq += v.x * v.x + v.y * v.y + v.z * v.z + v.w * v.w;
            if (ct == 0 && colIdx == 0) atomicAdd(&cnt[p], 1.0f);
        }
    }
    __syncthreads();

    // flush LDS partial tile -> global partial region (plain stores, no atomics)
    float* mysp = spart + ((size_t)ct * RB + rb) * (PNUM * COLS);
    for (int i = t; i < PNUM * COLS; i += TPB_B) mysp[i] = part[i];
    if (ct == 0) {
        float* mycp = cpart + (size_t)rb * PNUM;
        for (int i = t; i < PNUM; i += TPB_B) mycp[i] = cnt[i];
    }

    // block-reduce sum of squares, one f64 atomic per block
    #pragma unroll
    for (int off = 16; off; off >>= 1) ssq += __shfl_down(ssq, off, 32);
    if ((t & 31) == 0) red[t >> 5] = ssq;
    __syncthreads();
    if (t < TPB_B / 32) {
        float v = red[t];
        #pragma unroll
        for (int off = 16; off; off >>= 1) v += __shfl_down(v, off, 32);
        if (t == 0) atomicAdd(sumsq, (double)v);
    }
}

// ---------------- kernel B2: reduce partials -> centroids, counts ----------------
__global__ __launch_bounds__(256) void pafa_reduce(
    const float* __restrict__ spart, const float* __restrict__ cpart,
    float* __restrict__ centroids, float* __restrict__ countsOut)
{
    const int idx = blockIdx.x * 256 + threadIdx.x;      // 0..PNUM*DDIM-1
    const int d = idx & (DDIM - 1);
    const int p = idx >> 8;
    const int ctile = d / COLS;
    const int cl    = d & (COLS - 1);

    float s = 0.0f;
    const float* base = spart + ((size_t)ctile * RB * PNUM + p) * COLS + cl;
    #pragma unroll
    for (int rb = 0; rb < RB; ++rb) s += base[(size_t)rb * PNUM * COLS];

    float c = 0.0f;
    #pragma unroll
    for (int rb = 0; rb < RB; ++rb) c += cpart[rb * PNUM + p];
    c = fmaxf(c, 1.0f);

    centroids[idx] = s / c;
    if (d == 0) countsOut[p] = c;
}

// ---------------- kernel C: Gram diagonals via V_WMMA_F32_16X16X4_F32 ----------------
// S1 = sum_p ||c_p||^2 ; S2 = sum_p counts_p * ||c_p||^2
// For B = A^T, the f32 16x16x4 A and B operand layouts are identical:
//   lanes 0-15 : row = lane,    {VGPR0,VGPR1} = C[row][k0+0], C[row][k0+1]
//   lanes 16-31: row = lane-16, {VGPR0,VGPR1} = C[row][k0+2], C[row][k0+3]
__global__ __launch_bounds__(32) void pafa_gram(
    const float* __restrict__ centroids, const float* __restrict__ counts,
    float* __restrict__ S1, float* __restrict__ S2)
{
    const int lane  = threadIdx.x;            // 0..31
    const int m     = lane & 15;
    const int khalf = lane >> 4;              // 0 or 1
    const int row   = blockIdx.x * 16 + m;    // 16-row tile
    const float w   = counts[row];

    v8f acc1 = {0,0,0,0,0,0,0,0};
    v8f acc2 = {0,0,0,0,0,0,0,0};
    const float* rp = centroids + (size_t)row * DDIM + 2 * khalf;

    #pragma unroll 4
    for (int k0 = 0; k0 < DDIM; k0 += 4) {
        v2f b = *(const v2f*)(rp + k0);
        v2f a = b * w;
        acc1 = __builtin_amdgcn_wmma_f32_16x16x4_f32(false, b, false, b, (short)0, acc1, false, false);
        acc2 = __builtin_amdgcn_wmma_f32_16x16x4_f32(false, a, false, b, (short)0, acc2, false, false);
    }

    // diagonal (i,i): i<8 -> lane i, vgpr i ; i>=8 -> lane i+16, vgpr i-8
    float d1 = 0.0f, d2 = 0.0f;
    int sel = -1;
    if (lane < 8) sel = lane;
    else if (lane >= 24) sel = lane - 24;
    #pragma unroll
    for (int j = 0; j < 8; ++j) if (sel == j) { d1 = acc1[j]; d2 = acc2[j]; }

    #pragma unroll
    for (int off = 16; off; off >>= 1) {
        d1 += __shfl_down(d1, off, 32);
        d2 += __shfl_down(d2, off, 32);
    }
    if (lane == 0) { atomicAdd(S1, d1); atomicAdd(S2, d2); }
}

// ---------------- kernel D: column sums + final scalar ----------------
__global__ __launch_bounds__(256) void pafa_final(
    const float* __restrict__ centroids, const float* __restrict__ S1p,
    const float* __restrict__ S2p, const double* __restrict__ sumsqp,
    float* __restrict__ out)
{
    __shared__ float tred[DDIM];
    const int d = threadIdx.x;
    float td = 0.0f;
    for (int p = 0; p < PNUM; ++p) td += centroids[p * DDIM + d];
    tred[d] = td * td;
    __syncthreads();
    for (int s = DDIM / 2; s; s >>= 1) {
        if (d < s) tred[d] += tred[d + s];
        __syncthreads();
    }
    if (d == 0) {
        const float T  = tred[0];                  // ||sum_p c_p||^2
        const float S1 = *S1p;
        const float S2 = *S2p;
        const double sumsq = *sumsqp;
        const float Pf = (float)PNUM;
        const float within  = (float)(sumsq - (double)S2);
        const float between = Pf * S1 - T;
        const float pcsl = within / (between + 1e-6f);
        const float gpal = S1 / Pf - T / (Pf * Pf);
        out[0] = 0.1f * pcsl + 0.1f * gpal;
    }
}

extern "C" void kernel_launch(void* const* d_in, const int* in_sizes, int n_in,
                              void* d_out, int out_size, void* d_ws, size_t ws_size,
                              hipStream_t stream) {
    const float* feat = (const float*)d_in[0];
    const int*   pid  = (const int*)d_in[1];
    (void)in_sizes; (void)n_in; (void)out_size; (void)ws_size;

    char* ws = (char*)d_ws;
    double* sumsq     = (double*)(ws + WS_SUMSQ);
    float*  S1        = (float*)(ws + WS_S1);
    float*  S2        = (float*)(ws + WS_S2);
    float*  counts    = (float*)(ws + WS_COUNTS);
    float*  centroids = (float*)(ws + WS_CENTROIDS);
    float*  cpart     = (float*)(ws + WS_CPART);
    float*  spart     = (float*)(ws + WS_SPART);
    float*  out       = (float*)d_out;

    pafa_init<<<1, 32, 0, stream>>>(sumsq, S1, S2);
    pafa_partial<<<dim3(RB, CT), TPB_B, 0, stream>>>(feat, pid, spart, cpart, sumsq);
    pafa_reduce<<<(PNUM * DDIM) / 256, 256, 0, stream>>>(spart, cpart, centroids, counts);
    pafa_gram<<<PNUM / 16, 32, 0, stream>>>(centroids, counts, S1, S2);
    pafa_final<<<1, DDIM, 0, stream>>>(centroids, S1, S2, sumsq, out);
}